// CustomConv2d_81827716923731
// MI455X (gfx1250) — compile-verified
//
#include <hip/hip_runtime.h>
#include <stdint.h>

typedef float v8f  __attribute__((ext_vector_type(8)));
typedef float v2f  __attribute__((ext_vector_type(2)));
typedef unsigned int u32x4 __attribute__((ext_vector_type(4)));
typedef int  i32x4 __attribute__((ext_vector_type(4)));
typedef int  i32x8 __attribute__((ext_vector_type(8)));

#define BATCH  16
#define CIN    32
#define COUT   32
#define HDIM   64
#define WDIM   64
#define TILE_W 16
#define TILE_H 8
#define HALO_W 18                       // TILE_W + 2
#define HALO_H 10                       // TILE_H + 2
#define N_EW   (16 * CIN * 9)           // 4608 float4 entries (o in [0,16), .zw = o+16)
#define N_LOG  (CIN * HALO_H * HALO_W)  // 5760 floats

__global__ __launch_bounds__(256) void powconv_kernel(
    const float* __restrict__ x, const float* __restrict__ wgt,
    const float* __restrict__ bias, const float* __restrict__ expn,
    float* __restrict__ out)
{
  extern __shared__ __align__(16) float smem[];
  float4* sEW  = (float4*)smem;          // 73728 B : [o<16][c][ij] = (e_o, w_o, e_{o+16}, w_{o+16})
  float*  sLog = smem + 4 * N_EW;        // 23040 B : [c][10][18] haloed log2 tile
  float*  sBias = sLog + N_LOG;          // 128 B

  const int tid = threadIdx.x;
  const int b  = blockIdx.z;
  const int h0 = blockIdx.y * TILE_H;
  const int w0 = blockIdx.x * TILE_W;
  const float* xb = x + (size_t)b * CIN * HDIM * WDIM;

  // ---- stage (exponent, weight) pairs for o and o+16 as float4 ----
  for (int idx = tid; idx < N_EW; idx += 256) {
    float4 t;
    t.x = expn[idx];        t.y = wgt[idx];
    t.z = expn[idx + N_EW]; t.w = wgt[idx + N_EW];
    sEW[idx] = t;
  }
  if (tid < COUT) sBias[tid] = bias[tid];

  // ---- stage haloed input tile: TDM for interior blocks, clamped loads at borders ----
  const bool interior = (h0 >= 1) && (h0 + TILE_H + 1 <= HDIM) &&
                        (w0 >= 1) && (w0 + TILE_W + 1 <= WDIM);
  if (interior) {
    if (tid < 32) {  // wave 0 issues one TDM 3-D tile descriptor (x=18, y=10, z=32)
      const uint64_t ga = (uint64_t)(uintptr_t)(xb + (h0 - 1) * WDIM + (w0 - 1));
      const unsigned ldsOff = (unsigned)__builtin_amdgcn_groupstaticsize() +
                              (unsigned)(4u * N_EW * sizeof(float));
      u32x4 g0; i32x8 g1; i32x4 g2; i32x4 g3;
      g0[0] = 1u;                                   // count=1, user descriptor
      g0[1] = ldsOff;                               // lds_addr (bytes)
      g0[2] = (unsigned)(ga & 0xffffffffu);         // global_addr[31:0]
      g0[3] = (unsigned)((ga >> 32) & 0x01ffffffu)  // global_addr[56:32]
              | 0x80000000u;                        // type=2 ("image")
      g1[0] = 2 << 16;                              // data_size = 4 bytes
      g1[1] = (int)(WDIM << 16);                    // tensor_dim0 = 64
      g1[2] = (int)(HDIM << 16);                    // tensor_dim1 = 64
      g1[3] = HALO_W << 16;                         // tile_dim0 = 18
      g1[4] = HALO_H | (CIN << 16);                 // tile_dim1 = 10, tile_dim2 = 32
      g1[5] = WDIM;                                 // tensor_dim0_stride = 64
      g1[6] = (int)(((HDIM * WDIM) & 0xffff) << 16);// tensor_dim1_stride lo16 = 4096
      g1[7] = (HDIM * WDIM) >> 16;                  // tensor_dim1_stride hi = 0
      g2[0] = CIN; g2[1] = 0; g2[2] = 0; g2[3] = 0; // tensor_dim2 = 32
      g3[0] = 0;  g3[1] = 0;  g3[2] = 0;  g3[3] = 0;
#if defined(__clang_major__) && __clang_major__ >= 23
      i32x8 gX = {0, 0, 0, 0, 0, 0, 0, 0};
      __builtin_amdgcn_tensor_load_to_lds(g0, g1, g2, g3, gX, 0);
#else
      __builtin_amdgcn_tensor_load_to_lds(g0, g1, g2, g3, 0);
#endif
      __builtin_amdgcn_s_wait_tensorcnt(0);
    }
  } else {
    for (int idx = tid; idx < N_LOG; idx += 256) {
      const int xx = idx % HALO_W;
      const int t2 = idx / HALO_W;
      const int rr = t2 % HALO_H;
      const int c  = t2 / HALO_H;
      const int gr = h0 - 1 + rr, gc = w0 - 1 + xx;
      float v = 0.0f;
      if (gr >= 0 && gr < HDIM && gc >= 0 && gc < WDIM)
        v = xb[(c * HDIM + gr) * WDIM + gc];
      sLog[idx] = v;
    }
  }
  __syncthreads();

  // ---- in-place log2 conversion; clamp -inf so e==0 still yields x^0 == 1 ----
  for (int idx = tid; idx < N_LOG; idx += 256)
    sLog[idx] = fmaxf(__builtin_amdgcn_logf(sLog[idx]), -1.0e30f);
  __syncthreads();

  // ---- main loop: 1 v_exp_f32 per (b,o,h,w,c,i,j) term ----
  const int wv    = tid >> 5;       // wave -> output row in tile
  const int lane  = tid & 31;
  const int n     = lane & 15;      // output column in tile (WMMA N)
  const int hi    = lane >> 4;
  const int r     = wv;
  const int obase = 8 * hi;         // WMMA C-layout: vgpr v holds M = v + 8*hi

  float aLo[8], aHi[8];
  #pragma unroll
  for (int v = 0; v < 8; ++v) { aLo[v] = 0.0f; aHi[v] = 0.0f; }

  #pragma unroll 1
  for (int ij = 0; ij < 9; ++ij) {
    const int i = ij / 3, j = ij % 3;
    const float*  lrow = sLog + (r + i) * HALO_W + (n + j);
    const float4* ewc  = sEW + ij;
    #pragma unroll 1
    for (int c = 0; c < CIN; ++c) {
      const float Lg = lrow[c * (HALO_H * HALO_W)];   // log2 of the patch value
      #pragma unroll
      for (int v = 0; v < 8; ++v) {
        const float4 ew = ewc[((obase + v) * CIN + c) * 9];
        aLo[v] += ew.y * __builtin_amdgcn_exp2f(ew.x * Lg);   // o = obase+v
        aHi[v] += ew.w * __builtin_amdgcn_exp2f(ew.z * Lg);   // o = obase+v+16
      }
    }
  }

  // ---- bias add through WMMA f32 16x16x4: D = A*B + C, A[:,0]=bias, B[0,:]=1 ----
  v8f cLo, cHi;
  #pragma unroll
  for (int v = 0; v < 8; ++v) { cLo[v] = aLo[v]; cHi[v] = aHi[v]; }
  v2f A0, A1, Bv;
  A0.x = (hi == 0) ? sBias[n]      : 0.0f;  A0.y = 0.0f;  // A 16x4: lanes0-15 K=0..1
  A1.x = (hi == 0) ? sBias[16 + n] : 0.0f;  A1.y = 0.0f;
  Bv.x = (hi == 0) ? 1.0f          : 0.0f;  Bv.y = 0.0f;  // B 4x16: row K=0 = ones
  cLo = __builtin_amdgcn_wmma_f32_16x16x4_f32(false, A0, false, Bv, (short)0, cLo, false, false);
  cHi = __builtin_amdgcn_wmma_f32_16x16x4_f32(false, A1, false, Bv, (short)0, cHi, false, false);

  // ---- store (lanes 0-15 write 16 consecutive floats per channel row) ----
  float* ob = out + (((size_t)b * COUT) * HDIM + (h0 + r)) * WDIM + (w0 + n);
  #pragma unroll
  for (int v = 0; v < 8; ++v) {
    const int m = obase + v;
    ob[(size_t)m        * HDIM * WDIM] = cLo[v];
    ob[(size_t)(m + 16) * HDIM * WDIM] = cHi[v];
  }
}

extern "C" void kernel_launch(void* const* d_in, const int* in_sizes, int n_in,
                              void* d_out, int out_size, void* d_ws, size_t ws_size,
                              hipStream_t stream) {
  const float* x    = (const float*)d_in[0];
  const float* wgt  = (const float*)d_in[1];
  const float* bias = (const float*)d_in[2];
  const float* expn = (const float*)d_in[3];
  float* out = (float*)d_out;

  dim3 grid(WDIM / TILE_W, HDIM / TILE_H, BATCH);   // 4 x 8 x 16 blocks
  dim3 block(256);
  size_t smem = (size_t)(4 * N_EW + N_LOG + COUT) * sizeof(float);  // 96,896 B
  powconv_kernel<<<grid, block, smem, stream>>>(x, wgt, bias, expn, out);
}